// LocalGameStateGNN_54202487275973
// MI455X (gfx1250) — compile-verified
//
#include <hip/hip_runtime.h>
#include <hip/hip_bf16.h>
#include <math.h>

// ---------------- problem constants ----------------
#define BB 16
#define NN 22
#define TT 128
#define NG (BB*TT)            // 2048 graphs
#define NNODE (NG*NN)         // 45056 nodes
#define KNBR 16
#define X3D 192
#define VPROJ 64
#define HID 128
#define D0P 72                // padded node feature dim for layer 1 (69 -> 72)
#define NEDGE ((float)(NNODE*KNBR))

typedef __attribute__((ext_vector_type(2))) float v2f;
typedef __attribute__((ext_vector_type(8))) float v8f;

__device__ inline v8f wmma4(v2f a, v2f b, v8f c) {
  // D = A(16x4 f32) * B(4x16 f32) + C(16x16 f32)
  return __builtin_amdgcn_wmma_f32_16x16x4_f32(
      /*neg_a=*/false, a, /*neg_b=*/false, b,
      /*c_mod=*/(short)0, c, /*reuse_a=*/false, /*reuse_b=*/false);
}

// ---------------- workspace layout (in floats) ----------------
#define OFF_H0   ((size_t)0)
#define SZ_H0    ((size_t)NNODE * D0P)                  // 3,244,032
#define OFF_H1   (OFF_H0 + SZ_H0)
#define SZ_H     ((size_t)NNODE * HID)                  // 5,767,168
#define OFF_H2   (OFF_H1 + SZ_H)
#define OFF_ADJ  (OFF_H2 + SZ_H)                        // ints
#define SZ_ADJ   ((size_t)NNODE * KNBR)
#define OFF_W1P  (OFF_ADJ + SZ_ADJ)
#define SZ_W1P   ((size_t)HID * 2 * D0P)                // 128*144
#define OFF_STAT (OFF_W1P + SZ_W1P)                     // 6 slots * 256
#define OFF_SB   (OFF_STAT + 6*256)                     // 6 slots * 256

// ---------------- small prep kernels ----------------
__global__ void zero_stats_kernel(float* stats) {
  for (int i = threadIdx.x; i < 6*256; i += blockDim.x) stats[i] = 0.f;
}

__global__ void pad_w1_kernel(const float* __restrict__ w1, float* __restrict__ w1p) {
  int i = blockIdx.x * blockDim.x + threadIdx.x;       // 128*144
  if (i >= HID * 2 * D0P) return;
  int c = i / (2*D0P), j = i % (2*D0P);
  float v = 0.f;
  if (j < 69)                 v = w1[c*138 + j];        // center part
  else if (j >= 72 && j < 141) v = w1[c*138 + (j - 3)]; // diff part
  w1p[i] = v;
}

__global__ void build_base_kernel(const float* __restrict__ pos,
                                  const float* __restrict__ vel,
                                  const int*   __restrict__ team,
                                  float* __restrict__ h0) {
  int idx = blockIdx.x * blockDim.x + threadIdx.x;
  if (idx >= NNODE) return;
  int n = idx % NN;
  int g = idx / NN;        // b*T + t
  int t = g % TT;
  int b = g / TT;
  size_t src = (((size_t)b*NN + n)*TT + t);
  float* h = h0 + (size_t)idx * D0P;
  h[0] = pos[src*2];  h[1] = pos[src*2+1];
  h[2] = vel[src*2];  h[3] = vel[src*2+1];
  h[4] = (float)team[src];
  h[69] = 0.f; h[70] = 0.f; h[71] = 0.f;               // zero pad
}

// phi = phi_x3d @ W_vis^T  -> scattered into h0[:,5:69]
__global__ __launch_bounds__(128)
void phi_gemm_kernel(const float* __restrict__ phi, const float* __restrict__ wv,
                     float* __restrict__ h0) {
  const int lane = threadIdx.x & 31;
  const int wave = threadIdx.x >> 5;
  const int tile = blockIdx.x * 4 + wave;              // 2816 tiles of 16 rows
  const int r0   = tile * 16;
  const int m16  = lane & 15;
  const int kh   = (lane >> 4) << 1;
  const float* arow = phi + (size_t)(r0 + m16) * X3D;

  v8f acc[4];
#pragma unroll
  for (int nt = 0; nt < 4; nt++)
#pragma unroll
    for (int r = 0; r < 8; r++) acc[nt][r] = 0.f;

  for (int k = 0; k < X3D; k += 4) {
    int kk = k + kh;
    v2f a; a.x = arow[kk]; a.y = arow[kk+1];
#pragma unroll
    for (int nt = 0; nt < 4; nt++) {
      const float* wp = wv + (size_t)(nt*16 + m16) * X3D + kk;
      v2f bvec; bvec.x = wp[0]; bvec.y = wp[1];
      acc[nt] = wmma4(a, bvec, acc[nt]);
    }
  }
#pragma unroll
  for (int r = 0; r < 8; r++) {
    int row = (lane < 16) ? r : (r + 8);
    int rg  = r0 + row;                                // (b*NN+n)*TT + t
    int t   = rg % TT;
    int bn  = rg / TT;
    int n   = bn % NN;
    int b   = bn / NN;
    size_t idx = ((size_t)b*TT + t) * NN + n;
#pragma unroll
    for (int nt = 0; nt < 4; nt++)
      h0[idx*D0P + 5 + nt*16 + m16] = acc[nt][r];
  }
}

// brute force kNN per graph; lane = node
__global__ void knn_kernel(const float* __restrict__ pos,
                           const unsigned char* __restrict__ mask,
                           int* __restrict__ adj) {
  const int lane = threadIdx.x & 31;
  const int wave = threadIdx.x >> 5;
  const int g = blockIdx.x * 8 + wave;                 // 2048 graphs
  __shared__ float px[8][NN], py[8][NN];
  __shared__ int   mk[8][NN];
  if (lane < NN) {
    int b = g / TT, t = g % TT;
    size_t p = (((size_t)b*NN + lane)*TT + t) * 2;
    px[wave][lane] = pos[p]; py[wave][lane] = pos[p+1];
    mk[wave][lane] = mask[b*NN + lane] ? 1 : 0;
  }
  __syncthreads();
  if (lane < NN) {
    float xi = px[wave][lane], yi = py[wave][lane];
    int mi = mk[wave][lane];
    unsigned chosen = 0;
    for (int kk = 0; kk < KNBR; kk++) {
      float bmin = 3.0e38f; int jmin = lane;
      for (int j = 0; j < NN; j++) {
        if ((chosen >> j) & 1u) continue;
        if (j == lane) continue;
        if (!mk[wave][j]) continue;
        float dx = xi - px[wave][j], dy = yi - py[wave][j];
        float d = dx*dx + dy*dy;
        if (d < bmin) { bmin = d; jmin = j; }
      }
      chosen |= 1u << jmin;
      int tgt = mi ? jmin : lane;
      adj[((size_t)g*NN + lane)*KNBR + kk] = g*NN + tgt;
    }
  }
}

__global__ void finalize_stats_kernel(const float* __restrict__ stats,
                                      const float* __restrict__ gg,
                                      const float* __restrict__ bb,
                                      float* __restrict__ sb, float inv_cnt) {
  int c = threadIdx.x;                                 // 128
  float mean = stats[c] * inv_cnt;
  float var  = stats[128+c] * inv_cnt - mean*mean;
  float sc = gg[c] * rsqrtf(var + 1e-5f);
  sb[c] = sc;
  sb[128+c] = bb[c] - mean * sc;
}

// ---------------- edge-conv pass kernel ----------------
// MODE 1: GEMM1 -> stats.   MODE 2: GEMM1,BN1,ReLU,GEMM2 -> stats.
// MODE 3: full chain + BN2 + ReLU + row-max -> h_out.
template<int DIN, int MODE>
__global__ __launch_bounds__(128)
void edge_pass(const float* __restrict__ h_in, const int* __restrict__ adj,
               const float* __restrict__ w1,   const float* __restrict__ sb1,
               const float* __restrict__ w2,   const float* __restrict__ sb2,
               float* __restrict__ stats,      float* __restrict__ h_out) {
  constexpr int D2 = 2 * DIN;
  __shared__ float z1[4][(MODE >= 2) ? (16*HID) : 1];
  __shared__ float lstats[256];
  const int lane = threadIdx.x & 31;
  const int wave = threadIdx.x >> 5;
  const int node = blockIdx.x * 4 + wave;              // exact: NNODE/4 blocks
  const int m16  = lane & 15;
  const int kh   = (lane >> 4) << 1;

  if constexpr (MODE == 1 || MODE == 2) {
    for (int i = threadIdx.x; i < 256; i += blockDim.x) lstats[i] = 0.f;
    __syncthreads();
  }

  const float* ctr = h_in + (size_t)node * DIN;
  const int nbr = adj[node*KNBR + m16];                // lane's fixed edge
  const float* nb = h_in + (size_t)nbr * DIN;

  v8f acc[8];
#pragma unroll
  for (int nt = 0; nt < 8; nt++)
#pragma unroll
    for (int r = 0; r < 8; r++) acc[nt][r] = 0.f;

  // GEMM1: E(16 x 2*DIN) @ w1^T(2*DIN x 128)
  for (int k = 0; k < D2; k += 4) {
    int kk = k + kh;
    v2f a;
    if (kk < DIN) { a.x = ctr[kk]; a.y = ctr[kk+1]; }
    else { int q = kk - DIN; a.x = nb[q] - ctr[q]; a.y = nb[q+1] - ctr[q+1]; }
#pragma unroll
    for (int nt = 0; nt < 8; nt++) {
      const float* wp = w1 + (size_t)(nt*16 + m16) * D2 + kk;
      v2f bvec; bvec.x = wp[0]; bvec.y = wp[1];
      acc[nt] = wmma4(a, bvec, acc[nt]);
    }
  }

  if constexpr (MODE == 1) {
#pragma unroll
    for (int nt = 0; nt < 8; nt++) {
      float s = 0.f, s2 = 0.f;
#pragma unroll
      for (int r = 0; r < 8; r++) { float v = acc[nt][r]; s += v; s2 += v*v; }
      int c = nt*16 + m16;
      atomicAdd(&lstats[c], s);
      atomicAdd(&lstats[128 + c], s2);
    }
    __syncthreads();
    for (int i = threadIdx.x; i < 256; i += blockDim.x) atomicAdd(&stats[i], lstats[i]);
    return;
  } else {
    // BN1 + ReLU -> z1 in LDS (per-wave 16x128 tile, in-order DS within wave)
#pragma unroll
    for (int nt = 0; nt < 8; nt++) {
      int c = nt*16 + m16;
      float sc = sb1[c], bi = sb1[128 + c];
#pragma unroll
      for (int r = 0; r < 8; r++) {
        int row = (lane < 16) ? r : (r + 8);
        float v = acc[nt][r] * sc + bi;
        z1[wave][row*HID + c] = v > 0.f ? v : 0.f;
      }
    }
    // GEMM2: z1(16x128) @ w2^T(128x128)
    v8f acc2[8];
#pragma unroll
    for (int nt = 0; nt < 8; nt++)
#pragma unroll
      for (int r = 0; r < 8; r++) acc2[nt][r] = 0.f;

    for (int k = 0; k < HID; k += 4) {
      int kk = k + kh;
      v2f a; a.x = z1[wave][m16*HID + kk]; a.y = z1[wave][m16*HID + kk + 1];
#pragma unroll
      for (int nt = 0; nt < 8; nt++) {
        const float* wp = w2 + (size_t)(nt*16 + m16) * HID + kk;
        v2f bvec; bvec.x = wp[0]; bvec.y = wp[1];
        acc2[nt] = wmma4(a, bvec, acc2[nt]);
      }
    }

    if constexpr (MODE == 2) {
#pragma unroll
      for (int nt = 0; nt < 8; nt++) {
        float s = 0.f, s2 = 0.f;
#pragma unroll
        for (int r = 0; r < 8; r++) { float v = acc2[nt][r]; s += v; s2 += v*v; }
        int c = nt*16 + m16;
        atomicAdd(&lstats[c], s);
        atomicAdd(&lstats[128 + c], s2);
      }
      __syncthreads();
      for (int i = threadIdx.x; i < 256; i += blockDim.x) atomicAdd(&stats[i], lstats[i]);
    } else { // MODE 3: BN2 + ReLU + max over 16 edges
#pragma unroll
      for (int nt = 0; nt < 8; nt++) {
        int c = nt*16 + m16;
        float sc = sb2[c], bi = sb2[128 + c];
        float m = 0.f;                                 // ReLU => max >= 0
#pragma unroll
        for (int r = 0; r < 8; r++) {
          float v = acc2[nt][r] * sc + bi;
          v = v > 0.f ? v : 0.f;
          m = v > m ? v : m;
        }
        m = fmaxf(m, __shfl_xor(m, 16, 32));           // fold edge rows 8..15
        if (lane < 16) h_out[(size_t)node*HID + c] = m;
      }
    }
  }
}

// final transpose: (b*T+t, n, c) -> (b, n, t, c)
__global__ void write_out_kernel(const float* __restrict__ h3, float* __restrict__ out) {
  size_t o = (size_t)blockIdx.x * blockDim.x + threadIdx.x;
  if (o >= (size_t)NNODE * HID) return;
  int c = (int)(o & (HID - 1));
  size_t r = o >> 7;                                   // (b*NN+n)*TT + t
  int t = (int)(r % TT);
  size_t bn = r / TT;
  int n = (int)(bn % NN);
  int b = (int)(bn / NN);
  size_t idx = ((size_t)b*TT + t) * NN + n;
  out[o] = h3[idx*HID + c];
}

// ---------------- launcher ----------------
extern "C" void kernel_launch(void* const* d_in, const int* in_sizes, int n_in,
                              void* d_out, int out_size, void* d_ws, size_t ws_size,
                              hipStream_t stream) {
  (void)in_sizes; (void)n_in; (void)out_size; (void)ws_size;
  const float* phi_x3d   = (const float*)d_in[0];
  const float* positions = (const float*)d_in[1];
  const float* velocities= (const float*)d_in[2];
  const int*   team      = (const int*)d_in[3];
  const unsigned char* mask = (const unsigned char*)d_in[4];
  const float* W_vis     = (const float*)d_in[5];
  const float *Lw1[3], *Lg1[3], *Lb1[3], *Lw2[3], *Lg2[3], *Lb2[3];
  for (int l = 0; l < 3; l++) {
    int base = 6 + l*6;
    Lw1[l] = (const float*)d_in[base+0];
    Lg1[l] = (const float*)d_in[base+1];
    Lb1[l] = (const float*)d_in[base+2];
    Lw2[l] = (const float*)d_in[base+3];
    Lg2[l] = (const float*)d_in[base+4];
    Lb2[l] = (const float*)d_in[base+5];
  }
  float* ws   = (float*)d_ws;
  float* h0   = ws + OFF_H0;
  float* h1   = ws + OFF_H1;
  float* h2   = ws + OFF_H2;
  int*   adjb = (int*)(ws + OFF_ADJ);
  float* w1p0 = ws + OFF_W1P;
  float* stats= ws + OFF_STAT;
  float* sb   = ws + OFF_SB;
  const float inv_cnt = 1.0f / NEDGE;

  zero_stats_kernel<<<1, 256, 0, stream>>>(stats);
  pad_w1_kernel<<<(HID*2*D0P + 255)/256, 256, 0, stream>>>(Lw1[0], w1p0);
  build_base_kernel<<<(NNODE + 255)/256, 256, 0, stream>>>(positions, velocities, team, h0);
  phi_gemm_kernel<<<(NNODE/16)/4, 128, 0, stream>>>(phi_x3d, W_vis, h0);
  knn_kernel<<<NG/8, 256, 0, stream>>>(positions, mask, adjb);

  const int EB = NNODE / 4;  // edge-pass blocks (4 waves/block, 1 node/wave)

  // ---- layer 1 (DIN = 72) : h0 -> h1
  edge_pass<D0P,1><<<EB,128,0,stream>>>(h0, adjb, w1p0, nullptr, nullptr, nullptr, stats+0*256, nullptr);
  finalize_stats_kernel<<<1,128,0,stream>>>(stats+0*256, Lg1[0], Lb1[0], sb+0*256, inv_cnt);
  edge_pass<D0P,2><<<EB,128,0,stream>>>(h0, adjb, w1p0, sb+0*256, Lw2[0], nullptr, stats+1*256, nullptr);
  finalize_stats_kernel<<<1,128,0,stream>>>(stats+1*256, Lg2[0], Lb2[0], sb+1*256, inv_cnt);
  edge_pass<D0P,3><<<EB,128,0,stream>>>(h0, adjb, w1p0, sb+0*256, Lw2[0], sb+1*256, nullptr, h1);

  // ---- layer 2 (DIN = 128) : h1 -> h2
  edge_pass<HID,1><<<EB,128,0,stream>>>(h1, adjb, Lw1[1], nullptr, nullptr, nullptr, stats+2*256, nullptr);
  finalize_stats_kernel<<<1,128,0,stream>>>(stats+2*256, Lg1[1], Lb1[1], sb+2*256, inv_cnt);
  edge_pass<HID,2><<<EB,128,0,stream>>>(h1, adjb, Lw1[1], sb+2*256, Lw2[1], nullptr, stats+3*256, nullptr);
  finalize_stats_kernel<<<1,128,0,stream>>>(stats+3*256, Lg2[1], Lb2[1], sb+3*256, inv_cnt);
  edge_pass<HID,3><<<EB,128,0,stream>>>(h1, adjb, Lw1[1], sb+2*256, Lw2[1], sb+3*256, nullptr, h2);

  // ---- layer 3 (DIN = 128) : h2 -> h1 (reuse, h1 no longer needed)
  edge_pass<HID,1><<<EB,128,0,stream>>>(h2, adjb, Lw1[2], nullptr, nullptr, nullptr, stats+4*256, nullptr);
  finalize_stats_kernel<<<1,128,0,stream>>>(stats+4*256, Lg1[2], Lb1[2], sb+4*256, inv_cnt);
  edge_pass<HID,2><<<EB,128,0,stream>>>(h2, adjb, Lw1[2], sb+4*256, Lw2[2], nullptr, stats+5*256, nullptr);
  finalize_stats_kernel<<<1,128,0,stream>>>(stats+5*256, Lg2[2], Lb2[2], sb+5*256, inv_cnt);
  edge_pass<HID,3><<<EB,128,0,stream>>>(h2, adjb, Lw1[2], sb+4*256, Lw2[2], sb+5*256, nullptr, h1);

  write_out_kernel<<<(unsigned)(((size_t)NNODE*HID + 255)/256), 256, 0, stream>>>(h1, (float*)d_out);
}